// Voxelizer_34419867910943
// MI455X (gfx1250) — compile-verified
//
#include <hip/hip_runtime.h>
#include <hip/hip_bf16.h>
#include <stdint.h>

// Problem constants (must match reference)
#define B_    8
#define N_    262144
#define M_    4096
#define DGRID 20
#define NVOX  8000                 // 20^3
#define NSEG  (B_ * NVOX)
#define INV_VOX 20.0f              // 1/0.05

typedef unsigned int u32x4 __attribute__((ext_vector_type(4)));
typedef int          i32x8 __attribute__((ext_vector_type(8)));
typedef int          i32x4 __attribute__((ext_vector_type(4)));

// ---------------------------------------------------------------------------
// TDM: 1-D async copy of n_f32 dwords from global -> LDS (byte offset).
// Descriptor layout per CDNA5 ISA ch.8 (group0: 128b, group1: 256b).
// amdgpu-toolchain (clang-23) 6-arg builtin form.
// Caller must be a single wave (scalar branch) and later s_wait_tensorcnt.
// ---------------------------------------------------------------------------
__device__ __forceinline__ void tdm_load_1d(const void* gsrc,
                                            unsigned lds_byte_off,
                                            unsigned n_f32) {
  unsigned long long ga = (unsigned long long)(uintptr_t)gsrc;
  u32x4 g0;
  g0.x = 1u;                                              // count=1, user mode
  g0.y = lds_byte_off;                                    // lds_addr [63:32]
  g0.z = (unsigned)(ga & 0xFFFFFFFFull);                  // global_addr lo
  g0.w = (unsigned)((ga >> 32) & 0x01FFFFFFull)           // global_addr [56:32]
       | (2u << 30);                                      // type=2 ("image")
  i32x8 g1;
  g1[0] = (2 << 16);                                      // data_size=2 -> 4B
  g1[1] = (int)((n_f32 & 0xFFFFu) << 16);                 // tensor_dim0[15:0]
  g1[2] = (int)(((n_f32 >> 16) & 0xFFFFu) | (1u << 16));  // dim0 hi | tensor_dim1=1
  g1[3] = (int)((n_f32 & 0xFFFFu) << 16);                 // tile_dim0 = n_f32
  g1[4] = 0;                                              // tile_dim1/2 unused
  g1[5] = (int)n_f32;                                     // tensor_dim0_stride lo
  g1[6] = 0;
  g1[7] = 0;
  i32x4 gz4 = {0, 0, 0, 0};                               // groups 2/3 unused (<=2D)
  i32x8 gz8 = {0, 0, 0, 0, 0, 0, 0, 0};
  __builtin_amdgcn_tensor_load_to_lds(g0, g1, gz4, gz4, gz8, 0);
}

__device__ __forceinline__ int vox1(float p, float mn) {
  int i = (int)floorf((p - mn) * INV_VOX);
  return min(max(i, 0), DGRID - 1);
}

// ---------------------------------------------------------------------------
// Kernel 0: init workspace (min slots to +FLT_MAX bits, accumulators to 0)
// ---------------------------------------------------------------------------
__global__ void k_init(unsigned* __restrict__ minbits, float* __restrict__ acc) {
  int t = blockIdx.x * blockDim.x + threadIdx.x;
  if (t < B_ * 3) minbits[t] = 0x7F7FFFFFu;               // +FLT_MAX
  int stride = gridDim.x * blockDim.x;
  for (int i = t; i < NSEG * 10; i += stride) acc[i] = 0.0f;
}

// ---------------------------------------------------------------------------
// Kernel 1: per-batch coordinate min (bit-pattern atomicMin, x >= 0)
// grid = B_*32 blocks, 256 threads; each block covers 8192 points
// ---------------------------------------------------------------------------
__global__ void k_min(const float* __restrict__ x, unsigned* __restrict__ minbits) {
  const int PER_BLK = 8192;
  int b    = blockIdx.x >> 5;
  int part = blockIdx.x & 31;
  const float* xb = x + ((size_t)b * N_ + (size_t)part * PER_BLK) * 3;
  float m0 = 3.4e38f, m1 = 3.4e38f, m2 = 3.4e38f;
  for (int i = threadIdx.x; i < PER_BLK; i += blockDim.x) {
    m0 = fminf(m0, xb[i * 3 + 0]);
    m1 = fminf(m1, xb[i * 3 + 1]);
    m2 = fminf(m2, xb[i * 3 + 2]);
  }
  __shared__ unsigned sm[3];
  if (threadIdx.x < 3) sm[threadIdx.x] = 0x7F7FFFFFu;
  __syncthreads();
  atomicMin(&sm[0], __float_as_uint(m0));
  atomicMin(&sm[1], __float_as_uint(m1));
  atomicMin(&sm[2], __float_as_uint(m2));
  __syncthreads();
  if (threadIdx.x < 3) atomicMin(&minbits[b * 3 + threadIdx.x], sm[threadIdx.x]);
}

// ---------------------------------------------------------------------------
// Kernel 2 (pass A): count + first moments.
// LDS: hist 8000*4 f32 (128000B) + 2 x 8192-point staging buffers (2*98304B)
//    = 324608 B  (fits 320KB WGP LDS)
// grid = B_*8 blocks, 256 threads; block owns 32768 points of one batch.
// ---------------------------------------------------------------------------
__global__ void k_pass_a(const float* __restrict__ x,
                         const unsigned* __restrict__ minbits,
                         float* __restrict__ acc) {
  extern __shared__ float lds[];
  const int CHUNK = 8192, NCHUNK = 4, HIST = NVOX * 4;
  float* hist = lds;

  int b = blockIdx.x >> 3;
  int part = blockIdx.x & 7;
  size_t pbase = (size_t)b * N_ + (size_t)part * (CHUNK * NCHUNK);

  for (int i = threadIdx.x; i < HIST; i += blockDim.x) hist[i] = 0.0f;
  float mn0 = __uint_as_float(minbits[b * 3 + 0]);
  float mn1 = __uint_as_float(minbits[b * 3 + 1]);
  float mn2 = __uint_as_float(minbits[b * 3 + 2]);

  int wid = __builtin_amdgcn_readfirstlane((int)(threadIdx.x >> 5));
  if (wid == 0)   // prologue: DMA chunk 0 into buf0
    tdm_load_1d(x + pbase * 3, (unsigned)(HIST * sizeof(float)), CHUNK * 3);

  for (int c = 0; c < NCHUNK; ++c) {
    if (wid == 0) {
      if (c + 1 < NCHUNK) {
        unsigned off = (unsigned)((HIST + ((c + 1) & 1) * CHUNK * 3) * sizeof(float));
        tdm_load_1d(x + (pbase + (size_t)(c + 1) * CHUNK) * 3, off, CHUNK * 3);
        __builtin_amdgcn_s_wait_tensorcnt(1);   // chunk c complete
      } else {
        __builtin_amdgcn_s_wait_tensorcnt(0);
      }
    }
    __syncthreads();
    const float* s = lds + HIST + (c & 1) * (CHUNK * 3);
    for (int i = threadIdx.x; i < CHUNK; i += blockDim.x) {
      float px = s[i * 3 + 0], py = s[i * 3 + 1], pz = s[i * 3 + 2];
      int flat = (vox1(px, mn0) * DGRID + vox1(py, mn1)) * DGRID + vox1(pz, mn2);
      atomicAdd(&hist[flat * 4 + 0], 1.0f);
      atomicAdd(&hist[flat * 4 + 1], px);
      atomicAdd(&hist[flat * 4 + 2], py);
      atomicAdd(&hist[flat * 4 + 3], pz);
    }
    __syncthreads();
  }

  for (int v = threadIdx.x; v < NVOX; v += blockDim.x) {
    float cnt = hist[v * 4 + 0];
    if (cnt != 0.0f) {
      float* a = acc + ((size_t)b * NVOX + v) * 10;
      atomicAdd(a + 0, cnt);
      atomicAdd(a + 1, hist[v * 4 + 1]);
      atomicAdd(a + 2, hist[v * 4 + 2]);
      atomicAdd(a + 3, hist[v * 4 + 3]);
    }
  }
}

// ---------------------------------------------------------------------------
// Kernel 3 (pass B): 6 unique second moments (symmetric 3x3).
// LDS: hist 8000*6 f32 (192000B) + 2 x 4096-point staging (2*49152B)
//    = 290304 B
// ---------------------------------------------------------------------------
__global__ void k_pass_b(const float* __restrict__ x,
                         const unsigned* __restrict__ minbits,
                         float* __restrict__ acc) {
  extern __shared__ float lds[];
  const int CHUNK = 4096, NCHUNK = 8, HIST = NVOX * 6;
  float* hist = lds;

  int b = blockIdx.x >> 3;
  int part = blockIdx.x & 7;
  size_t pbase = (size_t)b * N_ + (size_t)part * (CHUNK * NCHUNK);

  for (int i = threadIdx.x; i < HIST; i += blockDim.x) hist[i] = 0.0f;
  float mn0 = __uint_as_float(minbits[b * 3 + 0]);
  float mn1 = __uint_as_float(minbits[b * 3 + 1]);
  float mn2 = __uint_as_float(minbits[b * 3 + 2]);

  int wid = __builtin_amdgcn_readfirstlane((int)(threadIdx.x >> 5));
  if (wid == 0)
    tdm_load_1d(x + pbase * 3, (unsigned)(HIST * sizeof(float)), CHUNK * 3);

  for (int c = 0; c < NCHUNK; ++c) {
    if (wid == 0) {
      if (c + 1 < NCHUNK) {
        unsigned off = (unsigned)((HIST + ((c + 1) & 1) * CHUNK * 3) * sizeof(float));
        tdm_load_1d(x + (pbase + (size_t)(c + 1) * CHUNK) * 3, off, CHUNK * 3);
        __builtin_amdgcn_s_wait_tensorcnt(1);
      } else {
        __builtin_amdgcn_s_wait_tensorcnt(0);
      }
    }
    __syncthreads();
    const float* s = lds + HIST + (c & 1) * (CHUNK * 3);
    for (int i = threadIdx.x; i < CHUNK; i += blockDim.x) {
      float px = s[i * 3 + 0], py = s[i * 3 + 1], pz = s[i * 3 + 2];
      int flat = (vox1(px, mn0) * DGRID + vox1(py, mn1)) * DGRID + vox1(pz, mn2);
      float* h = &hist[flat * 6];
      atomicAdd(h + 0, px * px);
      atomicAdd(h + 1, px * py);
      atomicAdd(h + 2, px * pz);
      atomicAdd(h + 3, py * py);
      atomicAdd(h + 4, py * pz);
      atomicAdd(h + 5, pz * pz);
    }
    __syncthreads();
  }

  for (int v = threadIdx.x; v < NVOX; v += blockDim.x) {
    float* a = acc + ((size_t)b * NVOX + v) * 10;
    #pragma unroll
    for (int k = 0; k < 6; ++k) atomicAdd(a + 4 + k, hist[v * 6 + k]);
  }
}

// ---------------------------------------------------------------------------
// Kernel 4: finalize mean/cov per voxel -> dists[NSEG][12]
// ---------------------------------------------------------------------------
__global__ void k_finalize(const float* __restrict__ acc, float* __restrict__ dists) {
  int v = blockIdx.x * blockDim.x + threadIdx.x;
  if (v >= NSEG) return;
  const float* a = acc + (size_t)v * 10;
  float inv = 1.0f / fmaxf(a[0], 1.0f);
  float mx = a[1] * inv, my = a[2] * inv, mz = a[3] * inv;
  float cxx = a[4] * inv - mx * mx;
  float cxy = a[5] * inv - mx * my;
  float cxz = a[6] * inv - mx * mz;
  float cyy = a[7] * inv - my * my;
  float cyz = a[8] * inv - my * mz;
  float czz = a[9] * inv - mz * mz;
  float* d = dists + (size_t)v * 12;
  d[0] = mx;  d[1] = my;  d[2] = mz;
  d[3] = cxx; d[4] = cxy; d[5] = cxz;
  d[6] = cxy; d[7] = cyy; d[8] = cyz;
  d[9] = cxz; d[10] = cyz; d[11] = czz;
}

// ---------------------------------------------------------------------------
// Kernel 5: gather sampled points -> out[B][M][12]
// ---------------------------------------------------------------------------
__global__ void k_gather(const float* __restrict__ x,
                         const long long* __restrict__ sidx,   // int64 indices
                         const unsigned* __restrict__ minbits,
                         const float* __restrict__ dists,
                         float* __restrict__ out) {
  int t = blockIdx.x * blockDim.x + threadIdx.x;
  if (t >= B_ * M_) return;
  int b = t / M_;
  long long idx = sidx[t];
  const float* p = x + ((size_t)b * N_ + (size_t)idx) * 3;
  float mn0 = __uint_as_float(minbits[b * 3 + 0]);
  float mn1 = __uint_as_float(minbits[b * 3 + 1]);
  float mn2 = __uint_as_float(minbits[b * 3 + 2]);
  int flat = (vox1(p[0], mn0) * DGRID + vox1(p[1], mn1)) * DGRID + vox1(p[2], mn2);
  const float* d = dists + ((size_t)b * NVOX + flat) * 12;
  float* o = out + (size_t)t * 12;
  #pragma unroll
  for (int k = 0; k < 12; ++k) o[k] = d[k];
}

// ---------------------------------------------------------------------------
extern "C" void kernel_launch(void* const* d_in, const int* in_sizes, int n_in,
                              void* d_out, int out_size, void* d_ws, size_t ws_size,
                              hipStream_t stream) {
  const float*     x    = (const float*)d_in[0];      // (B,N,3) f32
  const long long* sidx = (const long long*)d_in[1];  // (B,M) int64

  // Workspace layout: [minbits 24u | acc NSEG*10 f32 | dists NSEG*12 f32]
  unsigned* minbits = (unsigned*)d_ws;
  float*    acc     = (float*)((char*)d_ws + 128);
  float*    dists   = (float*)((char*)d_ws + 128 + (size_t)NSEG * 10 * sizeof(float));
  float*    out     = (float*)d_out;

  k_init<<<1024, 256, 0, stream>>>(minbits, acc);
  k_min<<<B_ * 32, 256, 0, stream>>>(x, minbits);

  size_t shA = (size_t)(NVOX * 4 + 2 * 8192 * 3) * sizeof(float);  // 324608 B
  size_t shB = (size_t)(NVOX * 6 + 2 * 4096 * 3) * sizeof(float);  // 290304 B
  k_pass_a<<<B_ * 8, 256, shA, stream>>>(x, minbits, acc);
  k_pass_b<<<B_ * 8, 256, shB, stream>>>(x, minbits, acc);

  k_finalize<<<(NSEG + 255) / 256, 256, 0, stream>>>(acc, dists);
  k_gather<<<(B_ * M_ + 255) / 256, 256, 0, stream>>>(x, sidx, minbits, dists, out);
}